// Net_2405181686361
// MI455X (gfx1250) — compile-verified
//
#include <hip/hip_runtime.h>
#include <hip/hip_bf16.h>

// Sizes from the reference
#define BB 8
#define NN 128
#define F0 32      // input features (layer1) == FOUT (layer2) == 32
#define SS 4
#define FOUT 32
#define NOUT 19
#define KDIM (5 * NN)          // 640 : k = s'*128 + j  (s'=4 is the bias slice)

typedef float v2f __attribute__((ext_vector_type(2)));
typedef float v8f __attribute__((ext_vector_type(8)));

// ---------------------------------------------------------------------------
// AE[b, i, k] with k = s'*128 + j :  s'<4 -> a[b,i,j]*e[b,i,j,s'] ; s'==4 -> a[b,i,j]
// Layer-invariant, built once.
// ---------------------------------------------------------------------------
__global__ void prep_ae(const float* __restrict__ a, const float* __restrict__ e,
                        float* __restrict__ AE) {
    int tid = blockIdx.x * blockDim.x + threadIdx.x;
    const int total = BB * NN * KDIM;
    if (tid >= total) return;
    int k  = tid % KDIM;
    int bi = tid / KDIM;           // b*128 + i
    int sp = k >> 7;               // 0..4
    int j  = k & 127;
    float av = a[bi * NN + j];
    float v  = (sp < 4) ? av * e[(size_t)(bi * NN + j) * SS + sp] : av;
    AE[tid] = v;
}

// ---------------------------------------------------------------------------
// Fused prep of Y and R via WMMA. All slices are M=128, K=32, N=32 GEMMs:
//   slice 0..3 : Y[b, s*128+j, o] = sum_f h[b,j,f] * w_kn[s, o*32+f]
//   slice 4    : Y[b, 4*128+j, o] = sum_f h[b,j,f] * bvec[o*32+f]
//   slice 5    : R[b, j, o]       = sum_f h[b,j,f] * root[f*32+o] + bias[o]
// One wave per 16x16 tile; 8 batches * 6 slices * 16 tiles = 768 waves.
// Slice selection is wave-uniform => EXEC stays all-1s around the WMMAs.
// ---------------------------------------------------------------------------
__global__ __launch_bounds__(128) void prep_yr_wmma(
        const float* __restrict__ h, int hstride,
        const float* __restrict__ w_kn, const float* __restrict__ bvec,
        const float* __restrict__ root, const float* __restrict__ bias,
        float* __restrict__ Y, float* __restrict__ R) {
    const int lane = threadIdx.x & 31;
    const int wave = threadIdx.x >> 5;
    const int flat = blockIdx.x * 4 + wave;       // 0..767
    const int b     = flat / 96;
    const int rem   = flat % 96;
    const int slice = rem >> 4;                   // 0..5
    const int t     = rem & 15;
    const int j0 = (t >> 1) * 16;                 // M tile origin (node j)
    const int o0 = (t & 1) * 16;                  // N tile origin (out feat)

    const int row = lane & 15;
    const int hi  = lane >> 4;
    const int col = lane & 15;

    // Wave-uniform B-matrix selection: element B[f, o] = base[f*fstr + o*ostr]
    const float* base;
    int fstr, ostr;
    if (slice < 4)       { base = w_kn + slice * (FOUT * F0); fstr = 1;    ostr = F0; }
    else if (slice == 4) { base = bvec;                        fstr = 1;    ostr = F0; }
    else                 { base = root;                        fstr = FOUT; ostr = 1;  }

    const float* aRow = h + (size_t)(b * NN + j0 + row) * hstride;
    const float* bCol = base + (size_t)(o0 + col) * ostr;

    v8f acc = {};
#pragma unroll
    for (int kk = 0; kk < F0; kk += 4) {
        const int kb = kk + 2 * hi;
        v2f afrag;
        afrag.x = aRow[kb];
        afrag.y = aRow[kb + 1];
        v2f bfrag;
        bfrag.x = bCol[(size_t)kb * fstr];
        bfrag.y = bCol[(size_t)(kb + 1) * fstr];
        acc = __builtin_amdgcn_wmma_f32_16x16x4_f32(
                false, afrag, false, bfrag, (short)0, acc, false, false);
    }

    if (slice < 5) {
#pragma unroll
        for (int r = 0; r < 8; ++r) {
            const int j = j0 + r + 8 * hi;
            Y[((size_t)b * KDIM + slice * NN + j) * FOUT + (o0 + col)] = acc[r];
        }
    } else {
        const float bv = bias[o0 + col];
#pragma unroll
        for (int r = 0; r < 8; ++r) {
            const int j = j0 + r + 8 * hi;
            R[((size_t)(b * NN + j)) * FOUT + (o0 + col)] = acc[r] + bv;
        }
    }
}

// ---------------------------------------------------------------------------
// Per batch b:  D(128x32) = AE(128x640) @ Y(640x32);  H = relu(D + R)
// One wave per 16x16 tile via V_WMMA_F32_16X16X4_F32 (exact fp32).
// Dual accumulators (even/odd K chunks) for ILP on the matrix pipe.
// 128 waves total: 8 batches * (8 M-tiles * 2 N-tiles). 4 waves / block.
// ---------------------------------------------------------------------------
__global__ __launch_bounds__(128) void ecc_wmma_gemm(
        const float* __restrict__ AE, const float* __restrict__ Y,
        const float* __restrict__ R,  float* __restrict__ Hout) {
    const int lane = threadIdx.x & 31;
    const int wave = threadIdx.x >> 5;
    const int flat = blockIdx.x * 4 + wave;   // 0..127
    const int b  = flat >> 4;
    const int t  = flat & 15;
    const int i0 = (t >> 1) * 16;             // M tile origin
    const int o0 = (t & 1) * 16;              // N tile origin

    const int row = lane & 15;                // A-row / B-col within tile
    const int hi  = lane >> 4;                // lane half selects K pair

    const float* aeRow = AE + (size_t)(b * NN + i0 + row) * KDIM;
    const float* yCol  = Y  + (size_t)b * (KDIM * FOUT) + (o0 + row);

    v8f acc0 = {};
    v8f acc1 = {};
#pragma unroll 2
    for (int kk = 0; kk < KDIM; kk += 8) {
        const int kb0 = kk + 2 * hi;          // even -> float2-aligned
        const int kb1 = kb0 + 4;
        v2f a0, a1, b0, b1;
        a0.x = aeRow[kb0];
        a0.y = aeRow[kb0 + 1];
        a1.x = aeRow[kb1];
        a1.y = aeRow[kb1 + 1];
        const float* yp0 = yCol + (size_t)kb0 * FOUT;
        const float* yp1 = yCol + (size_t)kb1 * FOUT;
        b0.x = yp0[0];
        b0.y = yp0[FOUT];
        b1.x = yp1[0];
        b1.y = yp1[FOUT];
        acc0 = __builtin_amdgcn_wmma_f32_16x16x4_f32(
                false, a0, false, b0, (short)0, acc0, false, false);
        acc1 = __builtin_amdgcn_wmma_f32_16x16x4_f32(
                false, a1, false, b1, (short)0, acc1, false, false);
    }
    v8f acc = acc0 + acc1;

    // Epilogue: D row = i0 + r + 8*hi, col = o0 + (lane&15)
    const int col = lane & 15;
#pragma unroll
    for (int r = 0; r < 8; ++r) {
        const int i = i0 + r + 8 * hi;
        const size_t idx = (size_t)(b * NN + i) * FOUT + (o0 + col);
        float v = acc[r] + R[idx];
        Hout[idx] = v > 0.f ? v : 0.f;
    }
}

// ---------------------------------------------------------------------------
// pooled[b,o] = sum_i h2[b,i,o] * mask[b,i];  out[b,t] = pooled @ w_dense + b_dense
// One block per batch (128 threads).
// ---------------------------------------------------------------------------
__global__ void pool_dense(const float* __restrict__ h2, const float* __restrict__ x,
                           const float* __restrict__ wd, const float* __restrict__ bd,
                           float* __restrict__ out) {
    __shared__ float part[4][FOUT];
    const int b = blockIdx.x;
    const int o = threadIdx.x & 31;
    const int g = threadIdx.x >> 5;           // 4 row groups
    float acc = 0.f;
    for (int i = g; i < NN; i += 4) {
        float m = x[(size_t)(b * NN + i) * (F0 + 1) + F0];   // mask channel
        acc += h2[(size_t)(b * NN + i) * FOUT + o] * m;
    }
    part[g][o] = acc;
    __syncthreads();
    if (threadIdx.x < NOUT) {
        const int tcol = threadIdx.x;
        float r = bd[tcol];
#pragma unroll
        for (int oo = 0; oo < FOUT; ++oo) {
            float p = part[0][oo] + part[1][oo] + part[2][oo] + part[3][oo];
            r += p * wd[oo * NOUT + tcol];
        }
        out[b * NOUT + tcol] = r;
    }
}

// ---------------------------------------------------------------------------
extern "C" void kernel_launch(void* const* d_in, const int* in_sizes, int n_in,
                              void* d_out, int out_size, void* d_ws, size_t ws_size,
                              hipStream_t stream) {
    const float* x       = (const float*)d_in[0];
    const float* a       = (const float*)d_in[1];
    const float* e       = (const float*)d_in[2];
    const float* w_kn1   = (const float*)d_in[3];
    const float* b_kn1   = (const float*)d_in[4];
    const float* root1   = (const float*)d_in[5];
    const float* bias1   = (const float*)d_in[6];
    const float* w_kn2   = (const float*)d_in[7];
    const float* b_kn2   = (const float*)d_in[8];
    const float* root2   = (const float*)d_in[9];
    const float* bias2   = (const float*)d_in[10];
    const float* w_dense = (const float*)d_in[11];
    const float* b_dense = (const float*)d_in[12];
    float* out = (float*)d_out;

    float* ws = (float*)d_ws;
    float* AE = ws;                                   // 8*128*640   = 655360 f
    float* Y  = AE + (size_t)BB * NN * KDIM;          // 8*640*32    = 163840 f
    float* R  = Y  + (size_t)BB * KDIM * FOUT;        // 8*128*32    =  32768 f
    float* H1 = R  + (size_t)BB * NN * FOUT;          //               32768 f
    float* H2 = H1 + (size_t)BB * NN * FOUT;          //               32768 f
    // total ~3.7 MB of workspace

    // Layer-invariant augmented edge matrix
    prep_ae<<<(BB * NN * KDIM + 255) / 256, 256, 0, stream>>>(a, e, AE);

    // Layer 1 (input = x[..., :32], row stride 33)
    prep_yr_wmma<<<192, 128, 0, stream>>>(x, F0 + 1, w_kn1, b_kn1, root1, bias1, Y, R);
    ecc_wmma_gemm<<<32, 128, 0, stream>>>(AE, Y, R, H1);

    // Layer 2 (input = H1, row stride 32)
    prep_yr_wmma<<<192, 128, 0, stream>>>(H1, FOUT, w_kn2, b_kn2, root2, bias2, Y, R);
    ecc_wmma_gemm<<<32, 128, 0, stream>>>(AE, Y, R, H2);

    // Masked pooling + dense head
    pool_dense<<<BB, 128, 0, stream>>>(H2, x, w_dense, b_dense, out);
}